// GCN_43705587204466
// MI455X (gfx1250) — compile-verified
//
#include <hip/hip_runtime.h>
#include <hip/hip_bf16.h>
#include <stdint.h>

// ---------------------------------------------------------------------------
// GCN forward for MI455X (gfx1250, wave32, WMMA + async-to-LDS).
//   h0   = X @ W1                      (WMMA f32 16x16x4, async LDS staging)
//   agg1 = spmm(h0)                    (COO, wave-per-edge, f32 atomics)
//   h1   = relu(agg1 + b1)
//   hw2  = h1 @ W2                     (WMMA f32 16x16x4)
//   agg2 = spmm(hw2)
//   out  = log_softmax(concat(agg2[index]+b2, s) @ Wl + bl)
// ---------------------------------------------------------------------------

#define NNODES 100000
#define NEDGES 1600000
#define NF     128          // NFEAT == NHID
#define NS     64
#define NCLASS 16
#define NB     8192
#define LDS_STRIDE 132      // 132 floats = 528 B (16B aligned, bank-conflict free)

typedef float v2f __attribute__((ext_vector_type(2)));
typedef float v8f __attribute__((ext_vector_type(8)));

// Async global->LDS copy of 16 bytes (GLOBAL_LOAD_ASYNC_TO_LDS_B128).
// LDS destination passed as low 32 bits of the generic pointer;
// global address as 64-bit VGPR pair; saddr = off.
__device__ __forceinline__ void async_copy_b128(float* lds_dst, const float* gsrc) {
  uint32_t ldsoff = (uint32_t)(uintptr_t)lds_dst;
  asm volatile("global_load_async_to_lds_b128 %0, %1, off"
               :: "v"(ldsoff), "v"(gsrc) : "memory");
}
__device__ __forceinline__ void wait_asynccnt0() {
  asm volatile("s_wait_asynccnt 0" ::: "memory");
}

// ---------------------------------------------------------------------------
// Dense GEMM  C[M x 128] = X[M x 128] * W[128 x 128]   (M multiple of 32)
// Block = 256 threads = 8 waves; block owns 32 rows of X staged in LDS via
// GLOBAL_LOAD_ASYNC_TO_LDS_B128. Wave w owns N-tile w and computes TWO 16x16
// M-tiles, reusing each B fragment for 2 WMMAs. K swept in steps of 4 with
// V_WMMA_F32_16X16X4_F32 (exact fp32, matches the fp32 reference).
// ---------------------------------------------------------------------------
__global__ __launch_bounds__(256)
void gcn_gemm128_wmma(const float* __restrict__ X,
                      const float* __restrict__ W,
                      float* __restrict__ C) {
  __shared__ float ldsX[32 * LDS_STRIDE];

  const int tid  = threadIdx.x;
  const int wave = tid >> 5;          // 0..7  -> N tile
  const int lane = tid & 31;
  const int m0   = blockIdx.x * 32;

  // async staged, coalesced load of the 32x128 X tile (float4 granularity)
  for (int i = tid; i < 32 * (NF / 4); i += 256) {
    const int r  = i >> 5;            // 0..31
    const int c4 = i & 31;            // 0..31 (float4 column)
    async_copy_b128(&ldsX[r * LDS_STRIDE + c4 * 4],
                    X + (size_t)(m0 + r) * NF + c4 * 4);
  }
  wait_asynccnt0();                   // my wave's async copies landed in LDS
  __syncthreads();                    // everyone's copies visible

  const int n0    = wave * 16;
  const int lhalf = lane >> 4;        // 0 or 1
  const int l15   = lane & 15;

  v8f acc0 = {}, acc1 = {};
#pragma unroll 4
  for (int k0 = 0; k0 < NF; k0 += 4) {
    const int ka = k0 + lhalf * 2;    // this lane-half's K pair
    v2f a0, a1, b;
    // A 16x4: lane&15 = M row; vgpr0/1 = K pair {0,1} (lanes 0-15) / {2,3}
    a0.x = ldsX[l15 * LDS_STRIDE + ka];
    a0.y = ldsX[l15 * LDS_STRIDE + ka + 1];
    a1.x = ldsX[(l15 + 16) * LDS_STRIDE + ka];
    a1.y = ldsX[(l15 + 16) * LDS_STRIDE + ka + 1];
    // B 4x16: N striped across lanes 0-15; K pair across vgpr0/1 + lane half
    b.x = W[(ka    ) * NF + n0 + l15];
    b.y = W[(ka + 1) * NF + n0 + l15];
    acc0 = __builtin_amdgcn_wmma_f32_16x16x4_f32(
        false, a0, false, b, (short)0, acc0, false, false);
    acc1 = __builtin_amdgcn_wmma_f32_16x16x4_f32(
        false, a1, false, b, (short)0, acc1, false, false);
  }

  // C/D layout: element i, lane L -> M = i + (L>>4)*8, N = L&15
  float* cp0 = C + (size_t)(m0 + lhalf * 8) * NF + n0 + l15;
  float* cp1 = cp0 + (size_t)16 * NF;
#pragma unroll
  for (int i = 0; i < 8; ++i) {
    cp0[(size_t)i * NF] = acc0[i];
    cp1[(size_t)i * NF] = acc1[i];
  }
}

// ---------------------------------------------------------------------------
// Zero-fill (float4 granularity)
// ---------------------------------------------------------------------------
__global__ __launch_bounds__(256)
void gcn_zero4(float4* __restrict__ p, int n4) {
  int i = blockIdx.x * blockDim.x + threadIdx.x;
  if (i < n4) p[i] = make_float4(0.f, 0.f, 0.f, 0.f);
}

// ---------------------------------------------------------------------------
// COO SpMM: Out[row[e], :] += val[e] * H[col[e], :]
// One wave per edge: 32 lanes x float4 = the whole 512B feature row,
// fully coalesced load (L2-resident: H is 51 MB vs 192 MB L2),
// scattered with global_atomic_add_f32.
// ---------------------------------------------------------------------------
__global__ __launch_bounds__(256)
void gcn_spmm_coo(const int* __restrict__ row,
                  const int* __restrict__ col,
                  const float* __restrict__ val,
                  const float* __restrict__ H,
                  float* __restrict__ Out) {
  const int lane   = threadIdx.x & 31;
  const int wave   = (blockIdx.x * blockDim.x + threadIdx.x) >> 5;
  const int nwaves = (gridDim.x * blockDim.x) >> 5;

  for (int e = wave; e < NEDGES; e += nwaves) {
    const int   r = row[e];
    const int   c = col[e];
    const float v = val[e];
    const float4 hv = ((const float4*)(H + (size_t)c * NF))[lane];
    float* dst = Out + (size_t)r * NF + lane * 4;
    atomicAdd(dst + 0, v * hv.x);
    atomicAdd(dst + 1, v * hv.y);
    atomicAdd(dst + 2, v * hv.z);
    atomicAdd(dst + 3, v * hv.w);
  }
}

// ---------------------------------------------------------------------------
// h1 = relu(agg + b1), vectorized float4 (bias period 128 floats = 32 float4)
// ---------------------------------------------------------------------------
__global__ __launch_bounds__(256)
void gcn_bias_relu(const float4* __restrict__ in,
                   const float4* __restrict__ bias,   // 32 float4 = 128 floats
                   float4* __restrict__ out, int n4) {
  int i = blockIdx.x * blockDim.x + threadIdx.x;
  if (i >= n4) return;
  float4 a = in[i];
  float4 b = bias[i & 31];
  float4 o;
  o.x = fmaxf(a.x + b.x, 0.f);
  o.y = fmaxf(a.y + b.y, 0.f);
  o.z = fmaxf(a.z + b.z, 0.f);
  o.w = fmaxf(a.w + b.w, 0.f);
  out[i] = o;
}

// ---------------------------------------------------------------------------
// Head: z[b,c] = (concat(H2[index[b]]+b2, s[b]) . Wl[:,c]) + bl[c]
//        out   = log_softmax(z)
// 128 threads/block = 8 rows x 16 classes; Wl (12 KB) staged in LDS.
// Reduction over the 16 classes via __shfl_xor inside the wave32.
// ---------------------------------------------------------------------------
__global__ __launch_bounds__(128)
void gcn_head(const float* __restrict__ H2,
              const float* __restrict__ b2,
              const float* __restrict__ S,
              const int*   __restrict__ index,
              const float* __restrict__ Wl,   // (128+64) x 16
              const float* __restrict__ bl,
              float* __restrict__ out) {
  __shared__ float wl[(NF + NS) * NCLASS];    // 3072 floats = 12 KB

  const int tid = threadIdx.x;
  for (int i = tid; i < (NF + NS) * NCLASS; i += 128) wl[i] = Wl[i];
  __syncthreads();

  const int r = blockIdx.x * 8 + (tid >> 4);  // batch row
  const int c = tid & 15;                     // class
  const int node = index[r];
  const float* h = H2 + (size_t)node * NF;
  const float* s = S + (size_t)r * NS;

  float acc = bl[c];
#pragma unroll 4
  for (int k = 0; k < NF; ++k)
    acc += (h[k] + b2[k]) * wl[k * NCLASS + c];
#pragma unroll 4
  for (int k = 0; k < NS; ++k)
    acc += s[k] * wl[(NF + k) * NCLASS + c];

  // log_softmax over the 16 lanes of this row (aligned 16-lane group in wave32)
  float m = acc;
  for (int off = 8; off >= 1; off >>= 1)
    m = fmaxf(m, __shfl_xor(m, off, 32));
  float e = expf(acc - m);
  float sum = e;
  for (int off = 8; off >= 1; off >>= 1)
    sum += __shfl_xor(sum, off, 32);

  out[(size_t)r * NCLASS + c] = acc - m - logf(sum);
}

// ---------------------------------------------------------------------------
extern "C" void kernel_launch(void* const* d_in, const int* in_sizes, int n_in,
                              void* d_out, int out_size, void* d_ws, size_t ws_size,
                              hipStream_t stream) {
  const float* s     = (const float*)d_in[0];
  const float* x     = (const float*)d_in[1];
  const int*   row   = (const int*)  d_in[2];
  const int*   col   = (const int*)  d_in[3];
  const float* val   = (const float*)d_in[4];
  const int*   index = (const int*)  d_in[5];
  const float* W1    = (const float*)d_in[6];
  const float* b1    = (const float*)d_in[7];
  const float* W2    = (const float*)d_in[8];
  const float* b2    = (const float*)d_in[9];
  const float* Wl    = (const float*)d_in[10];
  const float* bl    = (const float*)d_in[11];
  float* out = (float*)d_out;

  const size_t BUF_BYTES = (size_t)NNODES * NF * sizeof(float); // 51.2 MB
  float* buf0 = (float*)d_ws;                                   // ping
  float* buf1 = (float*)((char*)d_ws + BUF_BYTES);              // pong
  const int n4 = NNODES * NF / 4;                               // 3.2M float4

  const int gemmBlocks = NNODES / 32;        // 3125
  const int zeroBlocks = (n4 + 255) / 256;   // 12500
  const int spmmBlocks = 4096;               // 32768 waves, grid-stride edges
  const int headBlocks = NB / 8;             // 1024

  // 1) buf0 = x @ W1
  gcn_gemm128_wmma<<<gemmBlocks, 256, 0, stream>>>(x, W1, buf0);
  // 2) buf1 = 0 ; buf1 += adj @ buf0
  gcn_zero4<<<zeroBlocks, 256, 0, stream>>>((float4*)buf1, n4);
  gcn_spmm_coo<<<spmmBlocks, 256, 0, stream>>>(row, col, val, buf0, buf1);
  // 3) buf0 = relu(buf1 + b1)
  gcn_bias_relu<<<zeroBlocks, 256, 0, stream>>>((const float4*)buf1,
                                                (const float4*)b1,
                                                (float4*)buf0, n4);
  // 4) buf1 = buf0 @ W2
  gcn_gemm128_wmma<<<gemmBlocks, 256, 0, stream>>>(buf0, W2, buf1);
  // 5) buf0 = 0 ; buf0 += adj @ buf1
  gcn_zero4<<<zeroBlocks, 256, 0, stream>>>((float4*)buf0, n4);
  gcn_spmm_coo<<<spmmBlocks, 256, 0, stream>>>(row, col, val, buf1, buf0);
  // 6) head: gather + b2, concat s, linear, log_softmax
  gcn_head<<<headBlocks, 128, 0, stream>>>(buf0, b2, s, index, Wl, bl, out);
}